// CuGraphSAGEConv_2413771620571
// MI455X (gfx1250) — compile-verified
//
#include <hip/hip_runtime.h>

typedef __attribute__((ext_vector_type(2))) float v2f;
typedef __attribute__((ext_vector_type(8))) float v8f;

#define D_IN   64
#define D_FEAT 128           // 2 * D_IN (concat)
#define D_OUT  64
#define LDS_STRIDE 132       // 128 + 4 pad: avoids 16-way bank conflicts, keeps 8B align

// One block handles a tile of 16 target nodes.
// Waves 0..15: mean-aggregate one node each into LDS feat tile [16 x 128].
// Waves 0..3 : WMMA f32 16x16x4 GEMM  out[16x64] = feat @ W^T + b.
__global__ __launch_bounds__(512) void sage_fused_kernel(
    const float* __restrict__ x,       // [N, 64]
    const int*   __restrict__ row,     // [E] source node per edge
    const int*   __restrict__ colptr,  // [N+1]
    const float* __restrict__ W,       // [64, 128] (torch [out, in])
    const float* __restrict__ bias,    // [64]
    float*       __restrict__ out,     // [N, 64]
    int n_nodes)
{
    __shared__ float sW[D_OUT * LDS_STRIDE];     // 64 x 128 (padded)  ~33.8 KB
    __shared__ float sFeat[16 * LDS_STRIDE];     // 16 x 128 (padded)  ~ 8.4 KB

    const int tid  = threadIdx.x;
    const int wave = tid >> 5;
    const int lane = tid & 31;
    const int tile = blockIdx.x << 4;            // first node of this tile

    // ---- Stage W into LDS (all 512 threads, 16 floats each) ----
    for (int idx = tid; idx < D_OUT * D_FEAT; idx += 512) {
        sW[(idx >> 7) * LDS_STRIDE + (idx & 127)] = W[idx];
    }

    // ---- Phase 1: mean aggregation, one wave per node ----
    const int node = tile + wave;
    if (node < n_nodes) {
        const int start = colptr[node];
        const int end   = colptr[node + 1];
        const float2* __restrict__ x2 = (const float2*)x;  // [N, 32] float2
        float2 acc = make_float2(0.f, 0.f);
        for (int e = start; e < end; ++e) {
            const int src = row[e];                         // uniform across wave
            const float2 v = x2[src * 32 + lane];           // 256B coalesced / edge
            acc.x += v.x;
            acc.y += v.y;
        }
        const float inv = 1.0f / fmaxf((float)(end - start), 1.0f);
        float* fRowW = &sFeat[wave * LDS_STRIDE];
        fRowW[lane * 2 + 0] = acc.x * inv;                  // agg -> feat[0:64]
        fRowW[lane * 2 + 1] = acc.y * inv;
        const float2 xv = x2[node * 32 + lane];             // root -> feat[64:128]
        fRowW[D_IN + lane * 2 + 0] = xv.x;
        fRowW[D_IN + lane * 2 + 1] = xv.y;
    }
    __syncthreads();

    // ---- Phase 2: WMMA GEMM, wave w -> outchans [16w, 16w+16) ----
    if (wave < 4) {
        const int obase = wave << 4;
        const int mn    = lane & 15;          // M index for A, N index for B
        const int khalf = (lane >> 4) << 1;   // lanes 16..31 hold K+2,K+3

        const float* __restrict__ fRow = &sFeat[mn * LDS_STRIDE];
        const float* __restrict__ wRow = &sW[(obase + mn) * LDS_STRIDE];

        v8f c = {};   // accumulator starts at 0; bias added at the end
        #pragma unroll
        for (int k = 0; k < D_FEAT; k += 4) {
            const int kk = k + khalf;
            v2f a;  a[0] = fRow[kk];  a[1] = fRow[kk + 1];   // A: feat 16x4 slab
            v2f bb; bb[0] = wRow[kk]; bb[1] = wRow[kk + 1];  // B: W^T   4x16 slab
            // D = A x B + C   (f32, full precision)
            c = __builtin_amdgcn_wmma_f32_16x16x4_f32(
                    /*neg_a=*/false, a, /*neg_b=*/false, bb,
                    /*c_mod=*/(short)0, c, /*reuse_a=*/false, /*reuse_b=*/false);
        }

        const float bv  = bias[obase + mn];
        const int   col = obase + mn;
        const int   rb  = tile + ((lane >> 4) << 3);  // upper 16 lanes -> rows +8
        #pragma unroll
        for (int v = 0; v < 8; ++v) {
            const int r = rb + v;
            if (r < n_nodes) out[r * D_OUT + col] = c[v] + bv;  // 16 contiguous floats/row
        }
    }
}

extern "C" void kernel_launch(void* const* d_in, const int* in_sizes, int n_in,
                              void* d_out, int out_size, void* d_ws, size_t ws_size,
                              hipStream_t stream) {
    const float* x      = (const float*)d_in[0];
    const int*   row    = (const int*)  d_in[1];
    const int*   colptr = (const int*)  d_in[2];
    const float* W      = (const float*)d_in[3];
    const float* bias   = (const float*)d_in[4];
    float*       out    = (float*)d_out;

    const int n_nodes = in_sizes[0] / D_IN;           // x is [N, 64]
    const int blocks  = (n_nodes + 15) / 16;

    sage_fused_kernel<<<blocks, 512, 0, stream>>>(x, row, colptr, W, bias, out, n_nodes);
}